// CRFDecoder_9070970929274
// MI455X (gfx1250) — compile-verified
//
#include <hip/hip_runtime.h>
#include <math.h>

// Problem sizes (fixed by reference)
#define S_DIM 1024
#define B_DIM 64
#define H_DIM 512
#define V_DIM 256
#define NEGV  (-10000.0f)
#define BOS_T 1

typedef __attribute__((ext_vector_type(2))) float v2f;
typedef __attribute__((ext_vector_type(8))) float v8f;
typedef __attribute__((ext_vector_type(4))) unsigned int u32x4;
typedef __attribute__((ext_vector_type(4))) int i32x4;
typedef __attribute__((ext_vector_type(8))) int i32x8;

__device__ __forceinline__ v8f v8f_zero() {
    v8f z = {0.f, 0.f, 0.f, 0.f, 0.f, 0.f, 0.f, 0.f};
    return z;
}

// ---------------------------------------------------------------------------
// TDM: 2D dword tile (width x height, row stride = width) global -> LDS.
// D# built per CDNA5 ISA (group0: count/lds/global/type=2,
// group1: data_size=4B + dims/tile/stride, groups 2-3 zero => 2D).
// Issue once per wave; caller waits with s_wait_tensorcnt.
// ---------------------------------------------------------------------------
__device__ __forceinline__ void tdm_load_2d_dwords(const void* gsrc, unsigned lds_off,
                                                   unsigned width, unsigned height) {
    const unsigned long long ga = (unsigned long long)(uintptr_t)gsrc;
    u32x4 g0 = { 1u,                                   // count = 1 valid descriptor
                 lds_off,                              // lds_addr (bytes)
                 (unsigned)ga,                         // global_addr[31:0]
                 (unsigned)((ga >> 32) & 0x1FFFFFFu) | 0x80000000u }; // addr[56:32] | type=2
    i32x8 g1 = { (int)(2u << 16),                      // data_size = 2 (4 bytes); wg_mask = 0
                 (int)(width << 16),                   // tensor_dim0[15:0] @ bits 63:48
                 (int)(height << 16),                  // dim0 hi=0 | tensor_dim1[15:0] @ 95:80
                 (int)(width << 16),                   // dim1 hi=0 | tile_dim0 @ 127:112
                 (int)height,                          // tile_dim1 @ 143:128 ; tile_dim2 = 0
                 (int)width,                           // tensor_dim0_stride[31:0]
                 0, 0 };
    i32x4 gz = { 0, 0, 0, 0 };
#if defined(__clang_major__) && (__clang_major__ >= 23)
    i32x8 gz8 = {0,0,0,0,0,0,0,0};
    __builtin_amdgcn_tensor_load_to_lds(g0, g1, gz, gz, gz8, 0);
#else
    __builtin_amdgcn_tensor_load_to_lds(g0, g1, gz, gz, 0);
#endif
}

// ---------------------------------------------------------------------------
// Kernel 0: ET[(u/2)*512 + v*2 + (u&1)] = exp(trans[v][u])
//   (transposed for the scan GEMM B operand, K-pair interleaved so each WMMA
//    B fragment is one 8-byte LDS load). Coalesced reads over u.
// ---------------------------------------------------------------------------
__global__ __launch_bounds__(256) void et_exp_kernel(
    const float* __restrict__ trans, float* __restrict__ ETg)
{
    const int idx = blockIdx.x * 256 + threadIdx.x;   // 65536 total
    const int u = idx & 255, v = idx >> 8;
    ETg[(u >> 1) * 512 + v * 2 + (u & 1)] = __expf(trans[v * V_DIM + u]);
}

// ---------------------------------------------------------------------------
// Kernel 1: feats[m, v] = (enc[m, :] @ W[:, v] + b[v]) * mask(m)
//   M=65536, K=512, N=256.  Block: 256 thr (8 waves), tile 128(M) x 64(N).
//   Each wave: 16(M) x 64(N) via 4 f32 WMMA accumulators.
//   Bs is K-pair interleaved: Bs[(k/2)*160 + n*2 + (k&1)] -> b64 fragments.
// ---------------------------------------------------------------------------
__global__ __launch_bounds__(256) void feats_gemm_kernel(
    const float* __restrict__ enc, const float* __restrict__ W,
    const float* __restrict__ bias, const int* __restrict__ lengths,
    float* __restrict__ feats)
{
    __shared__ float As[128 * 36];   // 128 x 32 chunk, stride 36 (8B aligned, conflict-free frag reads)
    __shared__ float Bs[16 * 160];   // 16 k-pairs x 64 n x 2, stride 160 (halves hit disjoint banks)

    const int tid  = threadIdx.x;
    const int wave = tid >> 5;
    const int lane = tid & 31;
    const int half = lane >> 4;
    const int rl   = lane & 15;

    const int mb = blockIdx.x * 128;
    const int nb = blockIdx.y * 64;

    v8f c[4];
#pragma unroll
    for (int ti = 0; ti < 4; ++ti) c[ti] = v8f_zero();

    for (int kk = 0; kk < H_DIM; kk += 32) {
        // Stage A chunk: rows mb..mb+127, cols kk..kk+31 (coalesced float4)
        {
            const int r0 = tid >> 3, cg = tid & 7;
#pragma unroll
            for (int p = 0; p < 4; ++p) {
                const int r = r0 + p * 32;
                const float4 v = *(const float4*)(enc + (size_t)(mb + r) * H_DIM + kk + cg * 4);
                *(float4*)(&As[r * 36 + cg * 4]) = v;
                if (kk + 32 < H_DIM)  // prefetch next K chunk (global_prefetch_b8)
                    __builtin_prefetch(enc + (size_t)(mb + r) * H_DIM + kk + 32 + cg * 4, 0, 1);
            }
        }
        // Stage B chunk: W rows kk..kk+31, cols nb..nb+63 -> pair-interleaved
        {
            const int r0 = tid >> 4, cg = tid & 15;
#pragma unroll
            for (int p = 0; p < 2; ++p) {
                const int r = r0 + p * 16;
                const float4 v = *(const float4*)(W + (size_t)(kk + r) * V_DIM + nb + cg * 4);
                float* dst = &Bs[(r >> 1) * 160 + (r & 1)];
                dst[(cg * 4 + 0) * 2] = v.x;
                dst[(cg * 4 + 1) * 2] = v.y;
                dst[(cg * 4 + 2) * 2] = v.z;
                dst[(cg * 4 + 3) * 2] = v.w;
            }
        }
        __syncthreads();

#pragma unroll
        for (int k2 = 0; k2 < 32; k2 += 4) {
            // A fragment (16x4 f32): lanes 0-15 -> K = k2,k2+1 ; lanes 16-31 -> K = k2+2,k2+3
            const v2f a = *(const v2f*)&As[(wave * 16 + rl) * 36 + k2 + 2 * half];
#pragma unroll
            for (int ti = 0; ti < 4; ++ti) {
                const v2f bf = *(const v2f*)&Bs[((k2 >> 1) + half) * 160 + (ti * 16 + rl) * 2];
                c[ti] = __builtin_amdgcn_wmma_f32_16x16x4_f32(
                    false, a, false, bf, (short)0, c[ti], false, false);
            }
        }
        __syncthreads();
    }

    // Epilogue: + bias, * mask, store
#pragma unroll
    for (int ti = 0; ti < 4; ++ti) {
        const int col = nb + ti * 16 + rl;
        const float bv = bias[col];
#pragma unroll
        for (int j = 0; j < 8; ++j) {
            const int mrow = mb + wave * 16 + half * 8 + j;   // D layout: M = j + 8*half
            const int s  = mrow >> 6;
            const int bb = mrow & 63;
            const float msk = (s < lengths[bb]) ? 1.0f : 0.0f;
            feats[(size_t)mrow * V_DIM + col] = (c[ti][j] + bv) * msk;
        }
    }
}

// ---------------------------------------------------------------------------
// Kernel 2: sequential CRF scan.
//   4 workgroups x 256 threads; workgroup g owns batches b0=16g..16g+15.
//   ET (pair-interleaved exp(trans)^T, 256 KB) pulled into LDS once by the
//   Tensor Data Mover; resident for all 1024 steps (320 KB WGP LDS).
//   Per step: m = rowmax(Z); P = exp(Z-m); ACC = P @ ET (16x256x256 fp32 WMMA);
//             Z = mask ? feats + m + log(ACC) : Z.
// ---------------------------------------------------------------------------
#define ZP 258   // padded row stride for Zs / Ps (8B aligned b64 frag loads)

__global__ __launch_bounds__(256) void crf_scan_kernel(
    const float* __restrict__ feats, const float* __restrict__ ETg,
    const int* __restrict__ lengths, float* __restrict__ Zlse)
{
    extern __shared__ float smem[];
    float* ETs  = smem;                    // 65536 floats (256 KB), pair-interleaved
    float* Zs   = ETs + V_DIM * V_DIM;     // 16 x ZP
    float* Ps   = Zs + 16 * ZP;            // 16 x ZP
    float* pred = Ps + 16 * ZP;            // 16 x 16 reduction scratch
    float* mrow = pred + 256;              // 16 row maxes

    const int tid  = threadIdx.x;
    const int b0   = blockIdx.x * 16;
    const int i    = tid >> 4;             // local batch row 0..15
    const int l    = tid & 15;             // 16 helpers per row
    const int wave = tid >> 5;
    const int lane = tid & 31;
    const int half = lane >> 4;
    const int rl   = lane & 15;
    const int n0   = wave * 32;            // each wave covers 32 of the 256 v's

    // TDM: async DMA the whole ET matrix into LDS (one descriptor, wave 0 only)
    if (wave == 0) {
        tdm_load_2d_dwords(ETg, (unsigned)(uintptr_t)ETs, 512u, 128u);
        __builtin_amdgcn_s_wait_tensorcnt((short)0);
    }
    // Z0: NEG everywhere except BOS column
#pragma unroll
    for (int j = 0; j < 16; ++j) {
        const int v = l * 16 + j;
        Zs[i * ZP + v] = (v == BOS_T) ? 0.0f : NEGV;
    }
    int lenreg[8];
#pragma unroll
    for (int j = 0; j < 8; ++j) lenreg[j] = lengths[b0 + half * 8 + j];
    __syncthreads();

    for (int t = 0; t < S_DIM; ++t) {
        // 1) per-row max of Z
        float pm = -3.0e38f;
#pragma unroll
        for (int j = 0; j < 16; ++j) pm = fmaxf(pm, Zs[i * ZP + l * 16 + j]);
        pred[i * 16 + l] = pm;
        __syncthreads();
        if (l == 0) {
            float m = pred[i * 16];
#pragma unroll
            for (int j = 1; j < 16; ++j) m = fmaxf(m, pred[i * 16 + j]);
            mrow[i] = m;
        }
        __syncthreads();
        // 2) P = exp(Z - m)
        const float mi = mrow[i];
#pragma unroll
        for (int j = 0; j < 16; ++j) {
            const int v = l * 16 + j;
            Ps[i * ZP + v] = __expf(Zs[i * ZP + v] - mi);
        }
        __syncthreads();
        // 3) ACC = P (16x256) @ ET (256x256), this wave's 32 columns
        v8f c0 = v8f_zero(), c1 = v8f_zero();
        for (int kk = 0; kk < V_DIM; kk += 4) {
            const v2f a = *(const v2f*)&Ps[rl * ZP + kk + 2 * half];
            const int kp = ((kk >> 1) + half) * 512;
            const v2f bA = *(const v2f*)&ETs[kp + (n0 + rl) * 2];
            const v2f bB = *(const v2f*)&ETs[kp + (n0 + 16 + rl) * 2];
            c0 = __builtin_amdgcn_wmma_f32_16x16x4_f32(false, a, false, bA, (short)0, c0, false, false);
            c1 = __builtin_amdgcn_wmma_f32_16x16x4_f32(false, a, false, bB, (short)0, c1, false, false);
        }
        __syncthreads();
        // 4) Z update (masked)
#pragma unroll
        for (int j = 0; j < 8; ++j) {
            const int ib  = half * 8 + j;             // D layout row
            const float mv = mrow[ib];
            const bool msk = t < lenreg[j];
            {
                const int n = n0 + rl;
                const float zt = feats[((size_t)t * B_DIM + b0 + ib) * V_DIM + n] + mv + __logf(c0[j]);
                if (msk) Zs[ib * ZP + n] = zt;
            }
            {
                const int n = n0 + 16 + rl;
                const float zt = feats[((size_t)t * B_DIM + b0 + ib) * V_DIM + n] + mv + __logf(c1[j]);
                if (msk) Zs[ib * ZP + n] = zt;
            }
        }
        __syncthreads();
    }

    // Final logsumexp over v for each row
    float pm = -3.0e38f;
#pragma unroll
    for (int j = 0; j < 16; ++j) pm = fmaxf(pm, Zs[i * ZP + l * 16 + j]);
    pred[i * 16 + l] = pm;
    __syncthreads();
    if (l == 0) {
        float m = pred[i * 16];
#pragma unroll
        for (int j = 1; j < 16; ++j) m = fmaxf(m, pred[i * 16 + j]);
        mrow[i] = m;
    }
    __syncthreads();
    float ps = 0.0f;
#pragma unroll
    for (int j = 0; j < 16; ++j) ps += __expf(Zs[i * ZP + l * 16 + j] - mrow[i]);
    pred[i * 16 + l] = ps;
    __syncthreads();
    if (l == 0) {
        float ssum = 0.0f;
#pragma unroll
        for (int j = 0; j < 16; ++j) ssum += pred[i * 16 + j];
        Zlse[b0 + i] = mrow[i] + __logf(ssum);
    }
}

// ---------------------------------------------------------------------------
// Kernel 3: gold-path score per batch + final mean(Z - score)
// ---------------------------------------------------------------------------
__global__ __launch_bounds__(64) void crf_finalize_kernel(
    const float* __restrict__ feats, const float* __restrict__ trans,
    const int* __restrict__ targets, const int* __restrict__ lengths,
    const float* __restrict__ Zlse, float* __restrict__ out)
{
    __shared__ float red[B_DIM];
    const int b = threadIdx.x;
    const int len = lengths[b];
    float sc = 0.0f;
    int prev = BOS_T;
    for (int t = 0; t < S_DIM; ++t) {
        const int tgt = targets[t * B_DIM + b];
        const float msk = (t < len) ? 1.0f : 0.0f;
        // feats is already masked, so the emit term is correct as-is.
        sc += feats[((size_t)t * B_DIM + b) * V_DIM + tgt] + trans[tgt * V_DIM + prev] * msk;
        prev = tgt;
    }
    red[b] = Zlse[b] - sc;
    __syncthreads();
    if (b == 0) {
        float s = 0.0f;
        for (int k = 0; k < B_DIM; ++k) s += red[k];
        out[0] = s / (float)B_DIM;
    }
}

// ---------------------------------------------------------------------------
extern "C" void kernel_launch(void* const* d_in, const int* in_sizes, int n_in,
                              void* d_out, int out_size, void* d_ws, size_t ws_size,
                              hipStream_t stream)
{
    const float* enc     = (const float*)d_in[0];   // (S,B,H) f32
    const float* W       = (const float*)d_in[1];   // (H,V)   f32
    const float* bias    = (const float*)d_in[2];   // (V,)    f32
    const float* trans   = (const float*)d_in[3];   // (V,V)   f32
    const int*   targets = (const int*)d_in[4];     // (S,B)   int
    const int*   lengths = (const int*)d_in[5];     // (B,)    int
    float* out = (float*)d_out;

    // Workspace: feats (64 MB) + Zlse (64 floats) + ET (256 KB)
    float* feats = (float*)d_ws;
    float* Zlse  = feats + (size_t)S_DIM * B_DIM * V_DIM;
    float* ETg   = Zlse + B_DIM;

    et_exp_kernel<<<V_DIM * V_DIM / 256, 256, 0, stream>>>(trans, ETg);

    dim3 g1((S_DIM * B_DIM) / 128, V_DIM / 64);
    feats_gemm_kernel<<<g1, 256, 0, stream>>>(enc, W, bias, lengths, feats);

    const size_t smem_bytes =
        (size_t)(V_DIM * V_DIM + 2 * 16 * ZP + 256 + 16) * sizeof(float); // ~290 KB < 320 KB WGP LDS
    crf_scan_kernel<<<4, 256, smem_bytes, stream>>>(feats, ETg, lengths, Zlse);

    crf_finalize_kernel<<<1, 64, 0, stream>>>(feats, trans, targets, lengths, Zlse, out);
}